// RESUS_NN_2327872274812
// MI455X (gfx1250) — compile-verified
//
#include <hip/hip_runtime.h>
#include <hip/hip_bf16.h>
#include <math.h>

// Q=8192, S=30, D=512.
// out[0..Q)   = prediction
// out[Q..2Q)  = mean_l2

typedef __attribute__((ext_vector_type(2))) float v2f;
typedef __attribute__((ext_vector_type(8))) float v8f;

#define NQ 8192
#define NS 30
#define ND 512
#define THREADS 256
#define QPB 256   // queries per block (one per thread)
#define SGRP 5    // support rows per scheduling group (bounds SGPR live range)

// ---------------------------------------------------------------------------
// Tiny precompute: delta_y[s] = y[s] - sigmoid(pred[s]);  sn[s] = ||support_s||^2
// ws[0..29] = delta_y, ws[32..61] = sn
// ---------------------------------------------------------------------------
__global__ void resus_precompute(const float* __restrict__ support,
                                 const float* __restrict__ sup_y,
                                 const float* __restrict__ sup_pred,
                                 float* __restrict__ ws) {
  int s = threadIdx.x;
  if (s < NS) {
    const float* row = support + (size_t)s * ND;
    float acc = 0.f;
    for (int d = 0; d < ND; ++d) acc = fmaf(row[d], row[d], acc);
    float sig = 1.f / (1.f + expf(-sup_pred[s]));
    ws[s]      = sup_y[s] - sig;
    ws[32 + s] = acc;
  }
}

// ---------------------------------------------------------------------------
// Main kernel. Per thread: one query (L1 weighted score + softmax).
// Per wave: 32 queries as two 16-row WMMA tiles; Gram(q,s) via
// V_WMMA_F32_16X16X4_F32 against supports padded 30->32 columns.
// Pad columns 30,31: addresses are clamped to row 29 (no OOB reads); their
// Gram values are garbage-but-finite and are NEVER read back (finalization
// only consumes s < 30), so no masking/branching is needed at all.
// ---------------------------------------------------------------------------
__global__ __launch_bounds__(THREADS) void resus_main(
    const float* __restrict__ query,      // [NQ, ND]
    const float* __restrict__ support,    // [NS, ND]
    const float* __restrict__ qpred,      // [NQ]
    const float* __restrict__ w,          // [ND]
    const float* __restrict__ adj_scale,  // [30]
    const float* __restrict__ adj_bias,   // [30]
    const int*   __restrict__ nsamp,      // [1]
    const float* __restrict__ ws,         // [64] from precompute
    float* __restrict__ out)              // [2*NQ]
{
  __shared__ float lds_gram[QPB][32];     // 32 KB: gram rows per query

  const int tid    = threadIdx.x;
  const int lane   = tid & 31;
  const int wave   = tid >> 5;
  const int qblock = blockIdx.x * QPB;
  const int q      = qblock + tid;

  // -------- WMMA fragment addressing (32-bit A 16x4 / B 4x16 layouts) -----
  // A: VGPR0 = K={0|2}, VGPR1 = K={1|3}; lanes 0-15 -> K=0,1; lanes 16-31 -> K=2,3
  const int mrow  = lane & 15;
  const int khalf = (lane >> 4) << 1;       // 0 or 2
  const float* qA0 = query + (size_t)(qblock + wave * 32 + mrow) * ND + khalf;
  const float* qA1 = qA0 + (size_t)16 * ND;
  // B: lane -> column n; same khalf selection on K.
  const int ncol = lane & 15;
  const float* sB0 = support + (size_t)ncol * ND + khalf;          // n = 0..15
  const int n1  = 16 + ncol;                                       // n = 16..31
  const int n1c = (n1 < NS) ? n1 : (NS - 1);   // clamp pad lanes: no OOB reads
  const float* sB1 = support + (size_t)n1c * ND + khalf;

  v8f acc00 = {}; v8f acc01 = {}; v8f acc10 = {}; v8f acc11 = {};

  float score[NS];
#pragma unroll
  for (int s = 0; s < NS; ++s) score[s] = 0.f;
  float qn = 0.f;

  const float* qrow = query + (size_t)q * ND;

  for (int d0 = 0; d0 < ND; d0 += 8) {
    __builtin_prefetch(qrow + d0 + 64, 0, 3);   // global_prefetch_b8, stream query row

    // per-thread query chunk (vector loads)
    const float4 qa = *(const float4*)(qrow + d0);
    const float4 qb = *(const float4*)(qrow + d0 + 4);
    // uniform weight chunk (scalar loads -> SGPR operands)
    const float4 wa = *(const float4*)(w + d0);
    const float4 wb = *(const float4*)(w + d0 + 4);

    // Gram matrix: 2 k-steps of V_WMMA_F32_16X16X4_F32 — issue early so the
    // matrix pipe overlaps the VALU stream below.
#pragma unroll
    for (int kk = 0; kk < 8; kk += 4) {
      v2f a0; a0.x = qA0[d0 + kk]; a0.y = qA0[d0 + kk + 1];
      v2f a1; a1.x = qA1[d0 + kk]; a1.y = qA1[d0 + kk + 1];
      v2f b0; b0.x = sB0[d0 + kk]; b0.y = sB0[d0 + kk + 1];
      v2f b1; b1.x = sB1[d0 + kk]; b1.y = sB1[d0 + kk + 1];
      acc00 = __builtin_amdgcn_wmma_f32_16x16x4_f32(false, a0, false, b0,
                                                    (short)0, acc00, false, false);
      acc01 = __builtin_amdgcn_wmma_f32_16x16x4_f32(false, a0, false, b1,
                                                    (short)0, acc01, false, false);
      acc10 = __builtin_amdgcn_wmma_f32_16x16x4_f32(false, a1, false, b0,
                                                    (short)0, acc10, false, false);
      acc11 = __builtin_amdgcn_wmma_f32_16x16x4_f32(false, a1, false, b1,
                                                    (short)0, acc11, false, false);
    }

    qn = fmaf(qa.x, qa.x, qn); qn = fmaf(qa.y, qa.y, qn);
    qn = fmaf(qa.z, qa.z, qn); qn = fmaf(qa.w, qa.w, qn);
    qn = fmaf(qb.x, qb.x, qn); qn = fmaf(qb.y, qb.y, qn);
    qn = fmaf(qb.z, qb.z, qn); qn = fmaf(qb.w, qb.w, qn);

    // L1 weighted score: 2 VALU per (s,d) — v_sub (dual-issued) + v_fma with
    // |src| modifier, support/w values as SGPR operands.
    // Two-phase per group: batch-load SGRP rows (distinct SGPR octets, one
    // s_clause + one s_wait_kmcnt), then compute. The outer fence bounds the
    // live range to one group (prevents hoisting 240 scalars -> SGPR spills);
    // the inner fence pins the loads before the compute so all SGRP rows are
    // simultaneously live and the allocator can't recycle one SGPR octet
    // serially (per-row s_wait_kmcnt).
#pragma unroll
    for (int sg = 0; sg < NS; sg += SGRP) {
      asm volatile("" ::: "memory");    // group fence: bound SMEM live range
      float4 sa[SGRP], sb[SGRP];
#pragma unroll
      for (int i = 0; i < SGRP; ++i) {
        const float* srow = support + (size_t)(sg + i) * ND + d0; // uniform -> SMEM
        sa[i] = *(const float4*)(srow);
        sb[i] = *(const float4*)(srow + 4);
      }
      asm volatile("" ::: "memory");    // phase fence: loads stay batched
#pragma unroll
      for (int i = 0; i < SGRP; ++i) {
        const int s = sg + i;
        float sc = score[s];
        sc = fmaf(fabsf(qa.x - sa[i].x), wa.x, sc);
        sc = fmaf(fabsf(qa.y - sa[i].y), wa.y, sc);
        sc = fmaf(fabsf(qa.z - sa[i].z), wa.z, sc);
        sc = fmaf(fabsf(qa.w - sa[i].w), wa.w, sc);
        sc = fmaf(fabsf(qb.x - sb[i].x), wb.x, sc);
        sc = fmaf(fabsf(qb.y - sb[i].y), wb.y, sc);
        sc = fmaf(fabsf(qb.z - sb[i].z), wb.z, sc);
        sc = fmaf(fabsf(qb.w - sb[i].w), wb.w, sc);
        score[s] = sc;
      }
    }
  }

  // Spill WMMA C-layout -> per-query rows in LDS.
  // C 16x16 f32: VGPR r, lanes 0-15 -> M=r, N=lane; lanes 16-31 -> M=8+r, N=lane-16
  {
    const int half = lane >> 4;
#pragma unroll
    for (int r = 0; r < 8; ++r) {
      const int m = half * 8 + r;
      lds_gram[wave * 32 +      m][ncol     ] = acc00[r];
      lds_gram[wave * 32 +      m][16 + ncol] = acc01[r];
      lds_gram[wave * 32 + 16 + m][ncol     ] = acc10[r];
      lds_gram[wave * 32 + 16 + m][16 + ncol] = acc11[r];
    }
  }
  __syncthreads();

  // -------- Per-thread finalization --------
  // softmax over 30 supports + attention-weighted delta_y
  float mx = score[0];
#pragma unroll
  for (int s = 1; s < NS; ++s) mx = fmaxf(mx, score[s]);
  float esum = 0.f, dsum = 0.f;
#pragma unroll
  for (int s = 0; s < NS; ++s) {
    float e = expf(score[s] - mx);
    esum += e;
    dsum = fmaf(ws[s], e, dsum);          // ws[s] = delta_y[s] (uniform)
  }
  const float dyhat = dsum / esum;

  const int   ns    = nsamp[0];
  const float scale = fabsf(adj_scale[ns - 1]);
  const float bias  = adj_bias[ns - 1];
  const float pred  = fmaf(dyhat, scale, bias) + qpred[q];

  // mean L2: ||q-s||^2 = qn + sn - 2*gram
  float l2sum = 0.f;
#pragma unroll
  for (int s = 0; s < NS; ++s) {
    float g  = lds_gram[tid][s];
    float d2 = qn + ws[32 + s] - 2.f * g;
    d2 = fmaxf(d2, 0.f);
    l2sum += sqrtf(d2);
  }

  out[q]      = pred;
  out[NQ + q] = l2sum * (1.0f / (float)NS);
}

extern "C" void kernel_launch(void* const* d_in, const int* in_sizes, int n_in,
                              void* d_out, int out_size, void* d_ws, size_t ws_size,
                              hipStream_t stream) {
  const float* query    = (const float*)d_in[0];  // [8192,512]
  const float* support  = (const float*)d_in[1];  // [30,512]
  const float* sup_y    = (const float*)d_in[2];  // [30]
  const float* sup_pred = (const float*)d_in[3];  // [30]
  const float* qpred    = (const float*)d_in[4];  // [8192]
  const float* fc1_w    = (const float*)d_in[5];  // [512]
  // d_in[6] = fc1_b: softmax over s is invariant to a constant bias -> unused
  const float* adj_s    = (const float*)d_in[7];  // [30]
  const float* adj_b    = (const float*)d_in[8];  // [30]
  const int*   nsamp    = (const int*)d_in[9];    // [1] == 30

  float* out = (float*)d_out;
  float* ws  = (float*)d_ws;

  resus_precompute<<<1, 32, 0, stream>>>(support, sup_y, sup_pred, ws);
  resus_main<<<NQ / QPB, THREADS, 0, stream>>>(query, support, qpred, fc1_w,
                                               adj_s, adj_b, nsamp, ws, out);
}